// STM_71674414236237
// MI455X (gfx1250) — compile-verified
//
#include <hip/hip_runtime.h>

typedef float v2f __attribute__((ext_vector_type(2)));
typedef float v8f __attribute__((ext_vector_type(8)));

#define B_SZ   1024
#define D_SZ   512
#define MN     16384
#define NGRID  128
#define KC     64
#define LDSS   66   // padded LDS row stride (floats): 2-step bank walk, conflict-free b64 reads

// ---------------------------------------------------------------------------
// Kernel 0: row squared norms for x and weights; zero the output accumulator.
// One wave32 per row of 512 floats (16 floats/lane), shuffle reduction.
// ---------------------------------------------------------------------------
__global__ __launch_bounds__(256)
void som_norms_kernel(const float* __restrict__ x, const float* __restrict__ w,
                      float* __restrict__ x2, float* __restrict__ w2,
                      float* __restrict__ out) {
    if (blockIdx.x == 0 && threadIdx.x == 0) out[0] = 0.0f;
    const int wave = threadIdx.x >> 5;
    const int lane = threadIdx.x & 31;
    const int row  = blockIdx.x * 8 + wave;          // 0 .. B_SZ+MN-1
    const float* src; float* dst; int r;
    if (row < B_SZ) { src = x; dst = x2; r = row; }
    else            { src = w; dst = w2; r = row - B_SZ; }
    const float* p = src + (size_t)r * D_SZ + lane * 4;
    float s = 0.0f;
#pragma unroll
    for (int i = 0; i < 4; ++i) {
        float4 v = *(const float4*)(p + i * 128);
        s += v.x * v.x + v.y * v.y + v.z * v.z + v.w * v.w;
    }
#pragma unroll
    for (int off = 16; off > 0; off >>= 1) s += __shfl_xor(s, off, 32);
    if (lane == 0) dst[r] = s;
}

// ---------------------------------------------------------------------------
// Kernel 1: dists[b,n] = sqrt(max(x2[b] + w2[n] - 2 * x.w, 1e-12))
// Block: 256 threads = 8 waves, tile 32(M) x 64(N); wave -> one 16x16 tile.
// K staged through LDS in KC=64 chunks; V_WMMA_F32_16X16X4_F32 inner loop.
// ---------------------------------------------------------------------------
__global__ __launch_bounds__(256)
void som_dist_gemm_kernel(const float* __restrict__ x, const float* __restrict__ w,
                          const float* __restrict__ x2, const float* __restrict__ w2,
                          float* __restrict__ dists) {
    __shared__ float As[32 * LDSS];   // x rows   [32][KC]
    __shared__ float Bs[64 * LDSS];   // w rows   [64][KC]

    const int tid  = threadIdx.x;
    const int wv   = tid >> 5;        // 0..7
    const int lane = tid & 31;
    const int half = lane >> 4;       // K-pair select
    const int lm   = lane & 15;       // row (A) / col (B) within tile
    const int mi   = wv >> 2;         // 0..1
    const int ni   = wv & 3;          // 0..3

    const int bm = blockIdx.y * 32;
    const int bn = blockIdx.x * 64;

    v8f acc = {};

    for (int k0 = 0; k0 < D_SZ; k0 += KC) {
        __syncthreads();
        // stage As: 32 rows x 64 cols  (512 float4 loads, 2 per thread)
        for (int q = tid; q < 32 * 16; q += 256) {
            int row = q >> 4, c4 = (q & 15) << 2;
            float4 v = *(const float4*)(x + (size_t)(bm + row) * D_SZ + k0 + c4);
            float* d = &As[row * LDSS + c4];
            *(float2*)(d)     = make_float2(v.x, v.y);
            *(float2*)(d + 2) = make_float2(v.z, v.w);
        }
        // stage Bs: 64 rows x 64 cols  (1024 float4 loads, 4 per thread)
        for (int q = tid; q < 64 * 16; q += 256) {
            int row = q >> 4, c4 = (q & 15) << 2;
            float4 v = *(const float4*)(w + (size_t)(bn + row) * D_SZ + k0 + c4);
            float* d = &Bs[row * LDSS + c4];
            *(float2*)(d)     = make_float2(v.x, v.y);
            *(float2*)(d + 2) = make_float2(v.z, v.w);
        }
        __syncthreads();

        const float* ap = &As[(mi * 16 + lm) * LDSS + half * 2];
        const float* bp = &Bs[(ni * 16 + lm) * LDSS + half * 2];
#pragma unroll
        for (int kk = 0; kk < KC; kk += 4) {
            v2f a = *(const v2f*)(ap + kk);   // A[M=lm][k0+kk + 2*half + {0,1}]
            v2f b = *(const v2f*)(bp + kk);   // B[k0+kk + 2*half + {0,1}][N=lm]
            acc = __builtin_amdgcn_wmma_f32_16x16x4_f32(
                /*neg_a=*/false, a, /*neg_b=*/false, b,
                /*c_mod=*/(short)0, acc, /*reuse_a=*/false, /*reuse_b=*/false);
        }
    }

    // Epilogue: C/D layout -> VGPR r, lane half: M = r + 8*half, N = lm
    const int gn = bn + ni * 16 + lm;
    const float w2v = w2[gn];
#pragma unroll
    for (int r = 0; r < 8; ++r) {
        int gm = bm + mi * 16 + r + half * 8;
        float d2 = x2[gm] + w2v - 2.0f * acc[r];
        dists[(size_t)gm * MN + gn] = sqrtf(fmaxf(d2, 1e-12f));
    }
}

// ---------------------------------------------------------------------------
// Kernel 2: per-row argmin (first-index tie-break) -> BMU loc + target loc.
// ---------------------------------------------------------------------------
__global__ __launch_bounds__(256)
void som_argmin_kernel(const float* __restrict__ dists,
                       const int* __restrict__ labels,
                       const int* __restrict__ tpoints,
                       float* __restrict__ bi, float* __restrict__ bj,
                       float* __restrict__ ti, float* __restrict__ tj) {
    __shared__ float sv[256];
    __shared__ int   si[256];
    const int b = blockIdx.x;
    const float* row = dists + (size_t)b * MN;
    float best = 3.4e38f; int bidx = 0;
    for (int n = threadIdx.x; n < MN; n += 256) {
        float v = row[n];
        if (v < best) { best = v; bidx = n; }
    }
    sv[threadIdx.x] = best; si[threadIdx.x] = bidx;
    __syncthreads();
    for (int s = 128; s > 0; s >>= 1) {
        if (threadIdx.x < s) {
            float ov = sv[threadIdx.x + s]; int oi = si[threadIdx.x + s];
            if (ov < sv[threadIdx.x] || (ov == sv[threadIdx.x] && oi < si[threadIdx.x])) {
                sv[threadIdx.x] = ov; si[threadIdx.x] = oi;
            }
        }
        __syncthreads();
    }
    if (threadIdx.x == 0) {
        int idx = si[0];
        bi[b] = (float)(idx >> 7);            // row-major: i = idx / NGRID
        bj[b] = (float)(idx & (NGRID - 1));   // j = idx % NGRID
        int lab = labels[b];
        ti[b] = (float)tpoints[lab * 2 + 0];
        tj[b] = (float)tpoints[lab * 2 + 1];
    }
}

// ---------------------------------------------------------------------------
// Kernel 3: loss = 0.5 * sum_{b,n} dist * exp(-(d2_bmu + d2_targ) * inv_s2)
// Grid (MN/2048, B): each block sums 2048 entries of one batch row (L2 hits).
// ---------------------------------------------------------------------------
__global__ __launch_bounds__(256)
void som_loss_kernel(const float* __restrict__ dists,
                     const float* __restrict__ bi, const float* __restrict__ bj,
                     const float* __restrict__ ti, const float* __restrict__ tj,
                     const int* __restrict__ it, float* __restrict__ out) {
    __shared__ float red[256];
    const int b = blockIdx.y;
    const float bfi = bi[b], bfj = bj[b], tfi = ti[b], tfj = tj[b];
    const float sigma  = 64.0f * __expf(-(float)it[0] * 1.0e-3f);  // SIGMA0 * exp(-it/DECAY)
    const float inv_s2 = 1.0f / (sigma * sigma);
    const float* row = dists + (size_t)b * MN;
    float s = 0.0f;
    const int n0 = blockIdx.x * 2048 + threadIdx.x;
#pragma unroll
    for (int k = 0; k < 8; ++k) {
        int n = n0 + k * 256;
        float fi = (float)(n >> 7);
        float fj = (float)(n & (NGRID - 1));
        float d1 = (fi - bfi) * (fi - bfi) + (fj - bfj) * (fj - bfj);
        float d2 = (fi - tfi) * (fi - tfi) + (fj - tfj) * (fj - tfj);
        s += row[n] * __expf(-(d1 + d2) * inv_s2);
    }
    red[threadIdx.x] = s;
    __syncthreads();
    for (int t = 128; t > 0; t >>= 1) {
        if (threadIdx.x < t) red[threadIdx.x] += red[threadIdx.x + t];
        __syncthreads();
    }
    if (threadIdx.x == 0) atomicAdd(out, 0.5f * red[0]);
}

// ---------------------------------------------------------------------------
extern "C" void kernel_launch(void* const* d_in, const int* in_sizes, int n_in,
                              void* d_out, int out_size, void* d_ws, size_t ws_size,
                              hipStream_t stream) {
    const float* x       = (const float*)d_in[0];   // [B, D]
    const float* w       = (const float*)d_in[1];   // [MN, D]
    const int*   labels  = (const int*)d_in[2];     // [B]
    const int*   tpoints = (const int*)d_in[3];     // [10, 2]
    const int*   it      = (const int*)d_in[4];     // [1]
    float* out = (float*)d_out;

    float* ws    = (float*)d_ws;
    float* dists = ws;                               // B*MN floats (64 MB, fits L2)
    float* x2    = dists + (size_t)B_SZ * MN;        // B
    float* w2    = x2 + B_SZ;                        // MN
    float* bi    = w2 + MN;                          // B
    float* bj    = bi + B_SZ;                        // B
    float* ti    = bj + B_SZ;                        // B
    float* tj    = ti + B_SZ;                        // B

    som_norms_kernel<<<(B_SZ + MN) / 8, 256, 0, stream>>>(x, w, x2, w2, out);

    dim3 g1(MN / 64, B_SZ / 32);                     // 256 x 32 blocks
    som_dist_gemm_kernel<<<g1, 256, 0, stream>>>(x, w, x2, w2, dists);

    som_argmin_kernel<<<B_SZ, 256, 0, stream>>>(dists, labels, tpoints, bi, bj, ti, tj);

    dim3 g2(MN / 2048, B_SZ);
    som_loss_kernel<<<g2, 256, 0, stream>>>(dists, bi, bj, ti, tj, it, out);
}